// DecoderRNN_34368328303084
// MI455X (gfx1250) — compile-verified
//
#include <hip/hip_runtime.h>
#include <hip/hip_bf16.h>

// DecoderRNN single step on MI455X (gfx1250, wave32).
// Memory-bound fp32 pipeline: V_WMMA_F32_16X16X4_F32 for all GEMMs,
// 8 M-tile accumulators per wave so every weight byte is read once.

typedef float v2f __attribute__((ext_vector_type(2)));
typedef float v8f __attribute__((ext_vector_type(8)));

#define Bq   128
#define Hq   1024
#define Eq   512
#define Sq   512
#define Vq   32000

__device__ __forceinline__ float sigmoidf(float x) {
    return 1.0f / (1.0f + __expf(-x));
}

// C(128 x N) = act( [A0 | A1] @ [B0 | B1]^T + bias + bias2 )
//   A0: 128 x K0 (rows optionally gathered via `gather`), A1: 128 x K1 (may be null)
//   B0: N x ldb0 row-major (first K0 cols used), B1: N x ldb1 (first K1 cols)
// One wave per 16-wide N-slab; 8 accumulators cover all M=128 rows so each
// B fragment is loaded once per wave (critical for the 131MB W_dec pass).
__global__ __launch_bounds__(256) void gemm_f32_wmma(
    const float* __restrict__ A0, int K0,
    const float* __restrict__ B0, long long ldb0,
    const float* __restrict__ A1, int K1,
    const float* __restrict__ B1, long long ldb1,
    const float* __restrict__ bias, const float* __restrict__ bias2,
    const int*  __restrict__ gather,
    float* __restrict__ C, int N, int act)
{
    const int lane = threadIdx.x & 31;
    const int wave = threadIdx.x >> 5;
    const int ntile = blockIdx.x * 8 + wave;
    const int ncol = ntile * 16;
    const int lo = lane & 15;
    const int hi = lane >> 4;

    v8f acc[8];
#pragma unroll
    for (int mt = 0; mt < 8; ++mt)
#pragma unroll
        for (int e = 0; e < 8; ++e) acc[mt][e] = 0.0f;

    // Phase 0: A0 x B0 over K0
    {
        const float* a0p[8];
#pragma unroll
        for (int mt = 0; mt < 8; ++mt) {
            int m = mt * 16 + lo;
            int r = gather ? gather[m] : m;
            a0p[mt] = A0 + (long long)r * K0 + 2 * hi;
        }
        const float* bp = B0 + (long long)(ncol + lo) * ldb0 + 2 * hi;
        for (int k = 0; k < K0; k += 4) {
            v2f bf = *(const v2f*)(bp + k);
#pragma unroll
            for (int mt = 0; mt < 8; ++mt) {
                v2f af = *(const v2f*)(a0p[mt] + k);
                acc[mt] = __builtin_amdgcn_wmma_f32_16x16x4_f32(
                    false, af, false, bf, (short)0, acc[mt], false, false);
            }
        }
    }

    // Phase 1: A1 x B1 over K1 (second half of a K-concatenated input)
    if (A1) {
        const float* a1p[8];
#pragma unroll
        for (int mt = 0; mt < 8; ++mt)
            a1p[mt] = A1 + (long long)(mt * 16 + lo) * K1 + 2 * hi;
        const float* bp = B1 + (long long)(ncol + lo) * ldb1 + 2 * hi;
        for (int k = 0; k < K1; k += 4) {
            v2f bf = *(const v2f*)(bp + k);
#pragma unroll
            for (int mt = 0; mt < 8; ++mt) {
                v2f af = *(const v2f*)(a1p[mt] + k);
                acc[mt] = __builtin_amdgcn_wmma_f32_16x16x4_f32(
                    false, af, false, bf, (short)0, acc[mt], false, false);
            }
        }
    }

    float bv = 0.0f;
    if (bias)  bv += bias[ncol + lo];
    if (bias2) bv += bias2[ncol + lo];

    // C/D layout: VGPR r, lanes 0-15 -> M = mt*16 + r, lanes 16-31 -> M = mt*16 + r + 8
#pragma unroll
    for (int mt = 0; mt < 8; ++mt) {
#pragma unroll
        for (int r = 0; r < 8; ++r) {
            int m = mt * 16 + r + 8 * hi;
            float v = acc[mt][r] + bv;
            if (act == 1) v = tanhf(v);
            C[(long long)m * N + ncol + lo] = v;
        }
    }
}

// LSTM cell elementwise: gates (128 x 4096) -> h1, c1 (128 x 1024)
__global__ __launch_bounds__(256) void lstm_cell(
    const float* __restrict__ gates, const float* __restrict__ c0,
    float* __restrict__ h1, float* __restrict__ c1)
{
    int idx = blockIdx.x * blockDim.x + threadIdx.x;
    if (idx >= Bq * Hq) return;
    int m = idx >> 10;          // / 1024
    int j = idx & (Hq - 1);     // % 1024
    const float* g = gates + (long long)m * (4 * Hq);
    float iv = sigmoidf(g[j]);
    float fv = sigmoidf(g[Hq + j]);
    float gv = tanhf(g[2 * Hq + j]);
    float ov = sigmoidf(g[3 * Hq + j]);
    float cv = fv * c0[idx] + iv * gv;
    c1[idx] = cv;
    h1[idx] = ov * tanhf(cv);
}

// Fused attention per batch row: scores -> masked softmax -> weighted sum.
// ctx[b] (2MB) stays L2-hot between the two passes.
__global__ __launch_bounds__(256) void attention(
    const float* __restrict__ ctx, const unsigned char* __restrict__ mask,
    const float* __restrict__ target, float* __restrict__ alpha_out,
    float* __restrict__ weighted)
{
    const int b = blockIdx.x;
    const float* ctxb = ctx + (long long)b * Sq * Hq;
    const float* tgt  = target + (long long)b * Hq;
    __shared__ float sa[Sq];
    __shared__ float red[8];

    const int tid  = threadIdx.x;
    const int lane = tid & 31;
    const int wave = tid >> 5;

    // Keep target[b] in registers: lane owns h = lane + 32*t
    float treg[32];
#pragma unroll
    for (int t = 0; t < 32; ++t) treg[t] = tgt[lane + 32 * t];

    // Scores: one s per wave iteration, coalesced over h, wave-reduce
    for (int s = wave; s < Sq; s += 8) {
        const float* row = ctxb + (long long)s * Hq;
        float sum = 0.0f;
#pragma unroll
        for (int t = 0; t < 32; ++t) sum += row[lane + 32 * t] * treg[t];
#pragma unroll
        for (int off = 16; off; off >>= 1) sum += __shfl_xor(sum, off, 32);
        if (lane == 0) sa[s] = mask[b * Sq + s] ? -1.0e9f : sum;
    }
    __syncthreads();

    // Softmax over s=512 (block-wide)
    float mx = -3.0e38f;
    for (int s = tid; s < Sq; s += 256) mx = fmaxf(mx, sa[s]);
#pragma unroll
    for (int off = 16; off; off >>= 1) mx = fmaxf(mx, __shfl_xor(mx, off, 32));
    if (lane == 0) red[wave] = mx;
    __syncthreads();
    float bm = red[0];
#pragma unroll
    for (int w = 1; w < 8; ++w) bm = fmaxf(bm, red[w]);
    __syncthreads();

    float psum = 0.0f;
    for (int s = tid; s < Sq; s += 256) {
        float e = __expf(sa[s] - bm);
        sa[s] = e;
        psum += e;
    }
#pragma unroll
    for (int off = 16; off; off >>= 1) psum += __shfl_xor(psum, off, 32);
    if (lane == 0) red[wave] = psum;
    __syncthreads();
    float tot = 0.0f;
#pragma unroll
    for (int w = 0; w < 8; ++w) tot += red[w];
    float inv = 1.0f / tot;
    for (int s = tid; s < Sq; s += 256) {
        float a = sa[s] * inv;
        sa[s] = a;
        alpha_out[b * Sq + s] = a;
    }
    __syncthreads();

    // weighted[b,h] = sum_s alpha[s] * ctx[b,s,h]  (coalesced over h)
    for (int h = tid; h < Hq; h += 256) {
        float acc = 0.0f;
        for (int s = 0; s < Sq; ++s) acc += sa[s] * ctxb[(long long)s * Hq + h];
        weighted[(long long)b * Hq + h] = acc;
    }
}

extern "C" void kernel_launch(void* const* d_in, const int* in_sizes, int n_in,
                              void* d_out, int out_size, void* d_ws, size_t ws_size,
                              hipStream_t stream) {
    const int*   prev  = (const int*)d_in[0];
    const float* h0    = (const float*)d_in[1];
    const float* c0    = (const float*)d_in[2];
    const float* ctx   = (const float*)d_in[3];
    const unsigned char* cmask = (const unsigned char*)d_in[4];
    const float* emb   = (const float*)d_in[5];
    const float* W_ih  = (const float*)d_in[6];
    const float* W_hh  = (const float*)d_in[7];
    const float* b_ih  = (const float*)d_in[8];
    const float* b_hh  = (const float*)d_in[9];
    const float* W_in  = (const float*)d_in[10];
    const float* W_out = (const float*)d_in[11];
    const float* W_dec = (const float*)d_in[12];
    const float* b_dec = (const float*)d_in[13];

    float* out   = (float*)d_out;
    float* h1    = out;                       // 128*1024
    float* c1    = out + 131072;              // 128*1024
    float* alpha = out + 262144;              // 128*512
    float* logit = out + 327680;              // 128*32000

    float* ws       = (float*)d_ws;
    float* gates    = ws;                     // 128*4096
    float* target   = ws + 524288;            // 128*1024
    float* weighted = ws + 655360;            // 128*1024
    float* htilde   = ws + 786432;            // 128*1024

    // 1) gates = emb[prev] @ W_ih^T + h0 @ W_hh^T + b_ih + b_hh   (N=4096)
    gemm_f32_wmma<<<dim3(4096 / 16 / 8), 256, 0, stream>>>(
        emb, Eq, W_ih, Eq, h0, Hq, W_hh, Hq, b_ih, b_hh, prev, gates, 4 * Hq, 0);

    // 2) LSTM cell -> h1, c1
    lstm_cell<<<dim3((Bq * Hq) / 256), 256, 0, stream>>>(gates, c0, h1, c1);

    // 3) target = h1 @ W_in^T   (N=1024)
    gemm_f32_wmma<<<dim3(1024 / 16 / 8), 256, 0, stream>>>(
        h1, Hq, W_in, Hq, nullptr, 0, nullptr, 0, nullptr, nullptr, nullptr,
        target, Hq, 0);

    // 4) fused attention: scores + softmax + weighted sum
    attention<<<dim3(Bq), 256, 0, stream>>>(ctx, cmask, target, alpha, weighted);

    // 5) h_tilde = tanh([weighted | h1] @ W_out^T)   (W_out is 1024 x 2048)
    gemm_f32_wmma<<<dim3(1024 / 16 / 8), 256, 0, stream>>>(
        weighted, Hq, W_out, 2 * Hq, h1, Hq, W_out + Hq, 2 * Hq,
        nullptr, nullptr, nullptr, htilde, Hq, 1);

    // 6) logit = h_tilde @ W_dec^T + b_dec   (N=32000; W_dec read exactly once)
    gemm_f32_wmma<<<dim3(32000 / 16 / 8), 256, 0, stream>>>(
        htilde, Hq, W_dec, Hq, nullptr, 0, nullptr, 0, b_dec, nullptr, nullptr,
        logit, Vq, 0);
}